// RNN_69114613729260
// MI455X (gfx1250) — compile-verified
//
#include <hip/hip_runtime.h>

typedef __attribute__((ext_vector_type(2))) float v2f;
typedef __attribute__((ext_vector_type(8))) float v8f;

#define B_TOTAL 32768
#define T_SEQ   28
#define I_SZ    28
#define H_SZ    64
#define G_SZ    256   // 4*H
#define NCLS    10

// LDS pitches chosen so fragment reads (addr = row*pitch + 4*kk + 2*hi) are
// bank-conflict-free across 32 lanes (pitch % 32 == 4 -> 16 rows start in distinct 4-bank groups).
#define WIH_PITCH 36
#define WHH_PITCH 68
#define H_PITCH   68

#define WAVES_PER_WG  8
#define ROWS_PER_WAVE 16

// Fast activations on TRANS pipe (v_exp_f32 / v_rcp_f32 / v_tanh_f32),
// avoiding the v_div_fmas_f32 correctly-rounded division expansion.
__device__ __forceinline__ float fast_sigmoid(float x) {
    float e = __builtin_amdgcn_exp2f(x * -1.44269504088896f);  // e^-x
    return __builtin_amdgcn_rcpf(1.0f + e);
}
__device__ __forceinline__ float fast_tanh(float x) {
#if __has_builtin(__builtin_amdgcn_tanhf)
    return __builtin_amdgcn_tanhf(x);
#else
    float e = __builtin_amdgcn_exp2f(x * -2.88539008177793f);  // e^-2x
    return __builtin_amdgcn_rcpf(1.0f + e) * 2.0f - 1.0f;
#endif
}

__global__ __launch_bounds__(256)
void lstm_fused_kernel(const float* __restrict__ x,
                       const float* __restrict__ W_ih,
                       const float* __restrict__ W_hh,
                       const float* __restrict__ b_ih,
                       const float* __restrict__ b_hh,
                       const float* __restrict__ W_out,
                       const float* __restrict__ b_out,
                       float* __restrict__ out)
{
    __shared__ float sWih[G_SZ * WIH_PITCH];                       // ~36.9 KB
    __shared__ float sWhh[G_SZ * WHH_PITCH];                       // ~69.6 KB
    __shared__ float sH[WAVES_PER_WG * ROWS_PER_WAVE * H_PITCH];   // ~34.8 KB

    const int tid = threadIdx.x;

    // ---- Stage weights into LDS: thread t handles global row t (t in [0,256)) ----
    {
        const float4* srcHH = (const float4*)(W_hh + (size_t)tid * H_SZ);   // 64 f32
        #pragma unroll
        for (int q = 0; q < H_SZ / 4; ++q) {
            float4 v = srcHH[q];
            *(float4*)&sWhh[tid * WHH_PITCH + q * 4] = v;
        }
        const float4* srcIH = (const float4*)(W_ih + (size_t)tid * I_SZ);   // 28 f32
        #pragma unroll
        for (int q = 0; q < I_SZ / 4; ++q) {
            float4 v = srcIH[q];
            *(float4*)&sWih[tid * WIH_PITCH + q * 4] = v;
        }
    }
    __syncthreads();

    const int wave = tid >> 5;
    const int lane = tid & 31;
    const int col  = lane & 15;   // tile column (N) / A-matrix row (M) for this lane
    const int hi   = lane >> 4;   // K-half select in f32 fragments
    const int b0   = (blockIdx.x * WAVES_PER_WG + wave) * ROWS_PER_WAVE;

    float* hbuf = &sH[wave * ROWS_PER_WAVE * H_PITCH];

    // Bias per N-tile (depends only on this lane's column). Tile nt = gate*4 + j.
    float bias[16];
    #pragma unroll
    for (int nt = 0; nt < 16; ++nt) {
        int g = nt * 16 + col;
        bias[nt] = b_ih[g] + b_hh[g];
    }

    // Per-lane x base: A-fragment row for this lane is (b0 + col).
    const float* xlane = x + (size_t)(b0 + col) * (T_SEQ * I_SZ);

    v8f c_st[4];
    #pragma unroll
    for (int j = 0; j < 4; ++j) c_st[j] = (v8f){};

    for (int t = 0; t < T_SEQ; ++t) {
        // Opaque zero: keeps the weight-fragment LDS loads loop-variant so LICM
        // cannot hoist hundreds of invariant fragments and spill the accumulators.
        int zoff = 0;
        asm volatile("" : "+v"(zoff));

        // x A-fragments for this timestep, cached in regs, reused by all 4 groups.
        v2f xfrag[7];
        #pragma unroll
        for (int kk = 0; kk < 7; ++kk)
            xfrag[kk] = *(const v2f*)&xlane[t * I_SZ + kk * 4 + hi * 2];
        if (t + 1 < T_SEQ)
            __builtin_prefetch(&xlane[(t + 1) * I_SZ], 0, 3);  // global_prefetch

        v8f hnew[4];

        #pragma unroll
        for (int j = 0; j < 4; ++j) {          // hidden column group: cols j*16..j*16+15
            // Tiles {j, 4+j, 8+j, 12+j} = {i, f, g, o} for this group.
            v8f acc[4];
            #pragma unroll
            for (int g = 0; g < 4; ++g) {
                #pragma unroll
                for (int r = 0; r < 8; ++r) acc[g][r] = bias[g * 4 + j];
            }

            // gates += x_t[16x28] @ W_ih^T slice   (K = 28 = 7 k-steps of 4)
            #pragma unroll
            for (int kk = 0; kk < 7; ++kk) {
                const int k = kk * 4 + hi * 2;
                #pragma unroll
                for (int g = 0; g < 4; ++g) {
                    const int n = (g * 4 + j) * 16 + col;
                    v2f bfrag = *(const v2f*)&sWih[n * WIH_PITCH + k + zoff];
                    acc[g] = __builtin_amdgcn_wmma_f32_16x16x4_f32(
                        false, xfrag[kk], false, bfrag, (short)0, acc[g], false, false);
                }
            }

            // gates += h(t-1)[16x64] @ W_hh^T slice   (K = 64 = 16 k-steps)
            if (t > 0) {
                #pragma unroll
                for (int kk = 0; kk < 16; ++kk) {
                    const int k = kk * 4 + hi * 2;
                    v2f afrag = *(const v2f*)&hbuf[col * H_PITCH + k];
                    #pragma unroll
                    for (int g = 0; g < 4; ++g) {
                        const int n = (g * 4 + j) * 16 + col;
                        v2f bfrag = *(const v2f*)&sWhh[n * WHH_PITCH + k + zoff];
                        acc[g] = __builtin_amdgcn_wmma_f32_16x16x4_f32(
                            false, afrag, false, bfrag, (short)0, acc[g], false, false);
                    }
                }
            }

            // Cell update for this group; i/f/g/o are element-aligned across tiles.
            #pragma unroll
            for (int r = 0; r < 8; ++r) {
                float ig = fast_sigmoid(acc[0][r]);
                float fg = fast_sigmoid(acc[1][r]);
                float gg = fast_tanh   (acc[2][r]);
                float og = fast_sigmoid(acc[3][r]);
                float cn = fg * c_st[j][r] + ig * gg;
                c_st[j][r] = cn;
                hnew[j][r] = og * fast_tanh(cn);
            }
        }

        // Publish h(t) to LDS only after ALL reads of h(t-1) are done.
        #pragma unroll
        for (int j = 0; j < 4; ++j) {
            #pragma unroll
            for (int r = 0; r < 8; ++r)
                hbuf[(r + 8 * hi) * H_PITCH + j * 16 + col] = hnew[j][r];
        }
        // Next iteration's hbuf reads are same-wave LDS RAW -> s_wait_dscnt.
    }

    // ---- Head: out[16x10] = h_T @ W_out^T[64x10] + b_out (one 16x16 tile) ----
    {
        v8f acc;
        float bo = (col < NCLS) ? b_out[col] : 0.0f;
        #pragma unroll
        for (int r = 0; r < 8; ++r) acc[r] = bo;

        #pragma unroll
        for (int kk = 0; kk < 16; ++kk) {
            const int k = kk * 4 + hi * 2;
            v2f afrag = *(const v2f*)&hbuf[col * H_PITCH + k];
            v2f bfrag = {};
            if (col < NCLS) bfrag = *(const v2f*)&W_out[col * H_SZ + k];
            acc = __builtin_amdgcn_wmma_f32_16x16x4_f32(
                false, afrag, false, bfrag, (short)0, acc, false, false);
        }

        #pragma unroll
        for (int r = 0; r < 8; ++r) {
            const int row = r + 8 * hi;
            if (col < NCLS)
                out[(size_t)(b0 + row) * NCLS + col] = acc[r];
        }
    }
}

extern "C" void kernel_launch(void* const* d_in, const int* in_sizes, int n_in,
                              void* d_out, int out_size, void* d_ws, size_t ws_size,
                              hipStream_t stream) {
    const float* x     = (const float*)d_in[0];
    const float* W_ih  = (const float*)d_in[1];
    const float* W_hh  = (const float*)d_in[2];
    const float* b_ih  = (const float*)d_in[3];
    const float* b_hh  = (const float*)d_in[4];
    const float* W_out = (const float*)d_in[5];
    const float* b_out = (const float*)d_in[6];
    float* out = (float*)d_out;

    dim3 grid(B_TOTAL / (WAVES_PER_WG * ROWS_PER_WAVE));  // 256 workgroups
    dim3 block(WAVES_PER_WG * 32);                        // 8 wave32s
    hipLaunchKernelGGL(lstm_fused_kernel, grid, block, 0, stream,
                       x, W_ih, W_hh, b_ih, b_hh, W_out, b_out, out);
}